// PointGroup_62448824484062
// MI455X (gfx1250) — compile-verified
//
#include <hip/hip_runtime.h>
#include <hip/hip_bf16.h>

#define N_POINTS 500000
#define CDIM 64
#define NUM_CLASSES 20
#define N_PROP 128
#define M_ASSIGN 400000
#define PROPOSE_POINTS 100
#define BN_EPS 0.001f
#define EPSV 1e-8f

// workspace layout (float offsets)
#define WS_SUM     0      // [64]
#define WS_SUMSQ   64     // [64]
#define WS_SEGLOSS 128
#define WS_SEGCNT  129
#define WS_L1      130
#define WS_COS     131
#define WS_MASK    132
#define WS_SCALE   256    // [64]
#define WS_SHIFT   320    // [64]
#define WS_PN      384    // int[128]
#define WS_SCORE   512    // float[128]
#define WS_IP      640    // int[128]
#define WS_SEGPRED 768    // int[N_POINTS]
#define WS_PROBS   500992 // float[N_POINTS*NUM_CLASSES]

typedef __attribute__((ext_vector_type(2))) float v2f;
typedef __attribute__((ext_vector_type(8))) float v8f;

// ---- CDNA5 async global->LDS copy (16B per lane), tracked by ASYNCcnt ----
__device__ __forceinline__ void async_copy16(void* lds_ptr, const void* gptr) {
    const unsigned loff = (unsigned)(unsigned long long)lds_ptr;  // low 32 bits = LDS offset
    asm volatile("global_load_async_to_lds_b128 %0, %1, off"
                 :: "v"(loff), "v"(gptr)
                 : "memory");
}
__device__ __forceinline__ void wait_async0() {
    asm volatile("s_wait_asynccnt 0x0" ::: "memory");
}

// ---------------- k0: zero accumulators ----------------
__global__ void k0_init(float* ws) {
    for (int i = threadIdx.x; i < 768; i += blockDim.x) ws[i] = 0.0f;
}

// ---------------- k1: feat@w1 stats + feat@seg_w softmax ----------------
__global__ void __launch_bounds__(128)
k1_gemm_stats(const float* __restrict__ feat, const float* __restrict__ w1,
              const float* __restrict__ b1, const float* __restrict__ seg_w,
              const float* __restrict__ seg_b, const int* __restrict__ segment,
              float* __restrict__ ws) {
    __shared__ float s_w1[CDIM][CDIM];     // 16KB
    __shared__ float s_sw[CDIM][32];       // 8KB  (seg_w padded to 32 cols)
    __shared__ float s_feat[64][CDIM];     // 16KB
    __shared__ float s_logit[64][32];      // 8KB
    __shared__ float s_sum[CDIM], s_sumsq[CDIM];
    __shared__ float s_nll[2];

    const int t = threadIdx.x;
    const int rowBase = blockIdx.x * 64;

    // async-stage w1 (1024 x 16B chunks) and feat tile into LDS
    for (int i = 0; i < 8; ++i) {
        const int c = t + i * 128;
        async_copy16(&s_w1[0][0] + c * 4, w1 + c * 4);
    }
    const float* featTile = feat + (size_t)rowBase * CDIM;
    for (int i = 0; i < 8; ++i) {
        const int c = t + i * 128;          // 16B chunk id 0..1023
        const int r = c >> 4;
        const int col = (c & 15) * 4;
        if (rowBase + r < N_POINTS) {
            async_copy16(&s_feat[r][col], featTile + c * 4);
        } else {
            *(float4*)&s_feat[r][col] = make_float4(0.f, 0.f, 0.f, 0.f);
        }
    }
    if (t < CDIM) { s_sum[t] = 0.f; s_sumsq[t] = 0.f; }
    if (t < 2) s_nll[t] = 0.f;
    for (int i = t; i < CDIM * 32; i += 128) {
        const int k = i >> 5, n = i & 31;
        s_sw[k][n] = (n < NUM_CLASSES) ? seg_w[k * NUM_CLASSES + n] : 0.f;
    }
    wait_async0();
    __syncthreads();

    const int lane = t & 31, half = lane >> 4, l16 = lane & 15;
    const int wrow = (t >> 5) * 16;

    // h = feat@w1 + b1 : 4 column tiles of 16
    for (int nt = 0; nt < 4; ++nt) {
        const int ncol = nt * 16 + l16;
        v8f acc = {};
        for (int ks = 0; ks < 16; ++ks) {
            const int k0 = ks * 4 + 2 * half;
            v2f a, b;
            a.x = s_feat[wrow + l16][k0];
            a.y = s_feat[wrow + l16][k0 + 1];
            b.x = s_w1[k0][ncol];
            b.y = s_w1[k0 + 1][ncol];
            acc = __builtin_amdgcn_wmma_f32_16x16x4_f32(
                false, a, false, b, (short)0, acc, false, false);
        }
        const float bb = b1[ncol];
        float ls = 0.f, ls2 = 0.f;
        for (int j = 0; j < 8; ++j) {
            const int gr = rowBase + wrow + j + 8 * half;
            const float v = acc[j] + bb;
            if (gr < N_POINTS) { ls += v; ls2 += v * v; }
        }
        atomicAdd(&s_sum[ncol], ls);
        atomicAdd(&s_sumsq[ncol], ls2);
    }

    // logits = feat@seg_w + seg_b : 2 column tiles (cols 0..19 valid)
    for (int nt = 0; nt < 2; ++nt) {
        const int ncol = nt * 16 + l16;
        v8f acc = {};
        for (int ks = 0; ks < 16; ++ks) {
            const int k0 = ks * 4 + 2 * half;
            v2f a, b;
            a.x = s_feat[wrow + l16][k0];
            a.y = s_feat[wrow + l16][k0 + 1];
            b.x = s_sw[k0][ncol];
            b.y = s_sw[k0 + 1][ncol];
            acc = __builtin_amdgcn_wmma_f32_16x16x4_f32(
                false, a, false, b, (short)0, acc, false, false);
        }
        const float bb = (ncol < NUM_CLASSES) ? seg_b[ncol] : 0.f;
        for (int j = 0; j < 8; ++j)
            s_logit[wrow + j + 8 * half][ncol] = acc[j] + bb;
    }
    __syncthreads();

    // per-row softmax / argmax / NLL (threads 0..63, one row each)
    float* probs = ws + WS_PROBS;
    int* seg_pred = (int*)(ws + WS_SEGPRED);
    if (t < 64) {
        const int gr = rowBase + t;
        if (gr < N_POINTS) {
            float mx = -1e30f; int am = 0;
            for (int n = 0; n < NUM_CLASSES; ++n) {
                const float v = s_logit[t][n];
                if (v > mx) { mx = v; am = n; }
            }
            float se = 0.f;
            for (int n = 0; n < NUM_CLASSES; ++n) se += __expf(s_logit[t][n] - mx);
            const float inv = 1.f / se;
            float p[NUM_CLASSES];
            for (int n = 0; n < NUM_CLASSES; ++n) p[n] = __expf(s_logit[t][n] - mx) * inv;
            // 20 floats per row = 5 x 16B, row base 16B-aligned
            float4* dst = (float4*)(probs + (size_t)gr * NUM_CLASSES);
            for (int q = 0; q < 5; ++q)
                dst[q] = make_float4(p[q*4], p[q*4+1], p[q*4+2], p[q*4+3]);
            seg_pred[gr] = am;
            const int sg = segment[gr];
            if (sg != -1) {
                const int si = min(max(sg, 0), NUM_CLASSES - 1);
                const float nll = -(s_logit[t][si] - mx - __logf(se));
                atomicAdd(&s_nll[0], nll);
                atomicAdd(&s_nll[1], 1.f);
            }
        }
    }
    __syncthreads();
    if (t < CDIM) {
        atomicAdd(&ws[WS_SUM + t], s_sum[t]);
        atomicAdd(&ws[WS_SUMSQ + t], s_sumsq[t]);
    }
    if (t == 64) atomicAdd(&ws[WS_SEGLOSS], s_nll[0]);
    if (t == 65) atomicAdd(&ws[WS_SEGCNT], s_nll[1]);
}

// ---------------- k2: finalize BN scale/shift ----------------
__global__ void k2_bn(const float* __restrict__ gamma, const float* __restrict__ beta,
                      float* __restrict__ ws) {
    const int t = threadIdx.x;
    if (t < CDIM) {
        const float invN = 1.0f / (float)N_POINTS;
        const float mu = ws[WS_SUM + t] * invN;
        float var = ws[WS_SUMSQ + t] * invN - mu * mu;
        var = fmaxf(var, 0.f);
        const float sc = rsqrtf(var + BN_EPS) * gamma[t];
        ws[WS_SCALE + t] = sc;
        ws[WS_SHIFT + t] = beta[t] - mu * sc;
    }
}

// ---------------- k3: BN+ReLU h, bias losses ----------------
__global__ void __launch_bounds__(128)
k3_bias(const float* __restrict__ feat, const float* __restrict__ w1,
        const float* __restrict__ b1, const float* __restrict__ coord,
        const float* __restrict__ centroid, const float* __restrict__ w2,
        const float* __restrict__ b2, const int* __restrict__ instance,
        float* __restrict__ ws) {
    __shared__ float s_w1[CDIM][CDIM];   // 16KB
    __shared__ float s_feat[64][CDIM];   // 16KB
    __shared__ float s_h[64][CDIM];      // 16KB
    __shared__ float s_w2[CDIM][3];
    __shared__ float s_scale[CDIM], s_shift[CDIM];
    __shared__ float s_part[3];          // l1, cos, mask

    const int t = threadIdx.x;
    const int rowBase = blockIdx.x * 64;

    for (int i = 0; i < 8; ++i) {
        const int c = t + i * 128;
        async_copy16(&s_w1[0][0] + c * 4, w1 + c * 4);
    }
    const float* featTile = feat + (size_t)rowBase * CDIM;
    for (int i = 0; i < 8; ++i) {
        const int c = t + i * 128;
        const int r = c >> 4;
        const int col = (c & 15) * 4;
        if (rowBase + r < N_POINTS) {
            async_copy16(&s_feat[r][col], featTile + c * 4);
        } else {
            *(float4*)&s_feat[r][col] = make_float4(0.f, 0.f, 0.f, 0.f);
        }
    }
    if (t < 3) s_part[t] = 0.f;
    if (t < CDIM) { s_scale[t] = ws[WS_SCALE + t]; s_shift[t] = ws[WS_SHIFT + t]; }
    for (int i = t; i < CDIM * 3; i += 128) s_w2[i / 3][i % 3] = w2[i];
    wait_async0();
    __syncthreads();

    const int lane = t & 31, half = lane >> 4, l16 = lane & 15;
    const int wrow = (t >> 5) * 16;

    for (int nt = 0; nt < 4; ++nt) {
        const int ncol = nt * 16 + l16;
        v8f acc = {};
        for (int ks = 0; ks < 16; ++ks) {
            const int k0 = ks * 4 + 2 * half;
            v2f a, b;
            a.x = s_feat[wrow + l16][k0];
            a.y = s_feat[wrow + l16][k0 + 1];
            b.x = s_w1[k0][ncol];
            b.y = s_w1[k0 + 1][ncol];
            acc = __builtin_amdgcn_wmma_f32_16x16x4_f32(
                false, a, false, b, (short)0, acc, false, false);
        }
        const float bb = b1[ncol];
        const float sc = s_scale[ncol], sh = s_shift[ncol];
        for (int j = 0; j < 8; ++j) {
            const float hpre = acc[j] + bb;
            s_h[wrow + j + 8 * half][ncol] = fmaxf(0.f, hpre * sc + sh);
        }
    }
    __syncthreads();

    if (t < 64) {
        const int gr = rowBase + t;
        if (gr < N_POINTS) {
            float bp[3];
            for (int c = 0; c < 3; ++c) {
                float s = b2[c];
                for (int k = 0; k < CDIM; ++k) s += s_h[t][k] * s_w2[k][c];
                bp[c] = s;
            }
            float bg[3], dist = 0.f, dot = 0.f, n1 = 0.f, n2 = 0.f;
            for (int c = 0; c < 3; ++c) {
                bg[c] = centroid[gr * 3 + c] - coord[gr * 3 + c];
                dist += fabsf(bp[c] - bg[c]);
                dot += bp[c] * bg[c];
                n1 += bp[c] * bp[c];
                n2 += bg[c] * bg[c];
            }
            const float cosv = -(dot / ((sqrtf(n1) + EPSV) * (sqrtf(n2) + EPSV)));
            const float mask = (instance[gr] != -1) ? 1.f : 0.f;
            atomicAdd(&s_part[0], dist * mask);
            atomicAdd(&s_part[1], cosv * mask);
            atomicAdd(&s_part[2], mask);
        }
    }
    __syncthreads();
    if (t == 0) atomicAdd(&ws[WS_L1], s_part[0]);
    if (t == 1) atomicAdd(&ws[WS_COS], s_part[1]);
    if (t == 2) atomicAdd(&ws[WS_MASK], s_part[2]);
}

// ---------------- k4: proposal histogram ----------------
__global__ void k4_hist(const int* __restrict__ prop_ids, float* __restrict__ ws) {
    __shared__ int cnt[N_PROP];
    const int t = threadIdx.x;
    if (t < N_PROP) cnt[t] = 0;
    __syncthreads();
    const int stride = gridDim.x * blockDim.x;
    for (int i = blockIdx.x * blockDim.x + t; i < M_ASSIGN; i += stride)
        atomicAdd(&cnt[prop_ids[i]], 1);
    __syncthreads();
    int* pn = (int*)(ws + WS_PN);
    if (t < N_PROP && cnt[t] != 0) atomicAdd(&pn[t], cnt[t]);
}

// ---------------- k5: searchsorted + instance_pred ----------------
__global__ void k5_ipred(const int* __restrict__ prop_ids, const int* __restrict__ point_ids,
                         float* __restrict__ ws) {
    const int t = threadIdx.x;
    if (t < N_PROP) {
        int lo = 0, hi = M_ASSIGN;
        while (lo < hi) {
            const int mid = (lo + hi) >> 1;
            if (prop_ids[mid] < t) lo = mid + 1; else hi = mid;
        }
        const int first = min(lo, M_ASSIGN - 1);
        const int* seg_pred = (const int*)(ws + WS_SEGPRED);
        ((int*)(ws + WS_IP))[t] = seg_pred[point_ids[first]];
    }
}

// ---------------- k6: score gather-sum ----------------
__global__ void k6_score(const int* __restrict__ prop_ids, const int* __restrict__ point_ids,
                         float* __restrict__ ws) {
    __shared__ float sacc[N_PROP];
    const int t = threadIdx.x;
    if (t < N_PROP) sacc[t] = 0.f;
    __syncthreads();
    const float* probs = ws + WS_PROBS;
    const int* ip = (const int*)(ws + WS_IP);
    const int stride = gridDim.x * blockDim.x;
    for (int i = blockIdx.x * blockDim.x + t; i < M_ASSIGN; i += stride) {
        const int p = prop_ids[i];
        const int n = point_ids[i];
        atomicAdd(&sacc[p], probs[(size_t)n * NUM_CLASSES + ip[p]]);
    }
    __syncthreads();
    if (t < N_PROP && sacc[t] != 0.f) atomicAdd(&ws[WS_SCORE + t], sacc[t]);
}

// ---------------- k7: final outputs ----------------
__global__ void k7_final(const float* __restrict__ ws, float* __restrict__ out) {
    const int t = threadIdx.x;
    if (t == 0) {
        const float seg_loss = ws[WS_SEGLOSS] / fmaxf(ws[WS_SEGCNT], 1.0f);
        const float denom = ws[WS_MASK] + EPSV;
        out[0] = seg_loss + (ws[WS_L1] + ws[WS_COS]) / denom;
    }
    if (t < N_PROP) {
        const int pn = ((const int*)(ws + WS_PN))[t];
        const bool m = pn > PROPOSE_POINTS;
        out[1 + t] = m ? ws[WS_SCORE + t] / (float)max(pn, 1) : 0.0f;
        out[1 + N_PROP + t] = m ? 1.0f : 0.0f;
    }
}

extern "C" void kernel_launch(void* const* d_in, const int* in_sizes, int n_in,
                              void* d_out, int out_size, void* d_ws, size_t ws_size,
                              hipStream_t stream) {
    const float* feat     = (const float*)d_in[0];
    const float* coord    = (const float*)d_in[1];
    const float* centroid = (const float*)d_in[2];
    const float* w1       = (const float*)d_in[3];
    const float* b1       = (const float*)d_in[4];
    const float* gamma    = (const float*)d_in[5];
    const float* beta     = (const float*)d_in[6];
    const float* w2       = (const float*)d_in[7];
    const float* b2       = (const float*)d_in[8];
    const float* seg_w    = (const float*)d_in[9];
    const float* seg_b    = (const float*)d_in[10];
    const int*   segment  = (const int*)d_in[11];
    const int*   instance = (const int*)d_in[12];
    const int*   prop_ids = (const int*)d_in[13];
    const int*   point_ids= (const int*)d_in[14];
    float* ws  = (float*)d_ws;
    float* out = (float*)d_out;

    const int nTiles = (N_POINTS + 63) / 64;  // 7813

    k0_init<<<1, 256, 0, stream>>>(ws);
    k1_gemm_stats<<<nTiles, 128, 0, stream>>>(feat, w1, b1, seg_w, seg_b, segment, ws);
    k2_bn<<<1, 64, 0, stream>>>(gamma, beta, ws);
    k3_bias<<<nTiles, 128, 0, stream>>>(feat, w1, b1, coord, centroid, w2, b2, instance, ws);
    k4_hist<<<512, 256, 0, stream>>>(prop_ids, ws);
    k5_ipred<<<1, 128, 0, stream>>>(prop_ids, point_ids, ws);
    k6_score<<<512, 256, 0, stream>>>(prop_ids, point_ids, ws);
    k7_final<<<1, 256, 0, stream>>>(ws, out);
}